// CTCCRFNegativeLogLikelihood_18107582120298
// MI455X (gfx1250) — compile-verified
//
#include <hip/hip_runtime.h>

// CTC-CRF negative log likelihood forward for MI455X (gfx1250, wave32).
//
// Strategy (latency-bound sequential scan, ~1MB of data -> per-step latency,
// not bandwidth, is what matters):
//  * blocks 0..B-1 : numerator lattice, one block per batch, 256 threads = one
//    lattice column u, ping-pong LDS for the u-1 shift, 1 barrier per step.
//  * block B       : denominator for ALL 16 batches in ONE wave32 using
//    V_WMMA_F32_16X16X4_F32 in the exp-domain with per-batch max rescaling:
//      a'[b,s'] = log( sum_s exp(a[b,s]-m_b) * E[s,s'] ) + m_b + em[t,b,s']
//    with E = exp(transition) a constant 8x8 matrix. 16x8 @ 8x8 f32 matmul
//    = two chained WMMA 16x16x4 instructions per time step (K=0..3, K=4..7).
//  * tiny reduce kernel produces the scalar mean(logZ_den - logZ_num).
//
// Round-3 polish:
//  * emission loads indexed off the kernel-arg base with 32-bit running
//    offsets -> global_load (round 2's pointer array had regressed to
//    flat_load, which also ties up DScnt).
//  * all log/exp on guaranteed-normal-range values use raw v_log_f32 /
//    v_exp_f32 (base-2 builtins + ln2/log2e scaling) instead of the precise
//    libm lowering with denorm fixup branches (~8 instrs -> 2 per log).

typedef __attribute__((ext_vector_type(2))) float v2f;
typedef __attribute__((ext_vector_type(8))) float v8f;

#define NEGV (-1e30f)
#define LN2F 0.6931471805599453f
#define LOG2EF 1.4426950408889634f

__device__ __forceinline__ float fast_exp(float x) {
  // v_exp_f32; underflows cleanly to 0 for very negative x (NEG sentinel).
  return __builtin_amdgcn_exp2f(x * LOG2EF);
}
__device__ __forceinline__ float fast_log(float x) {
  // v_log_f32 * ln2; only used on values guaranteed normal (no denorm fixup).
  return __builtin_amdgcn_logf(x) * LN2F;
}

__device__ __forceinline__ float lae(float a, float b) {
  // logaddexp; matches jnp.logaddexp incl. the NEG sentinel:
  // both -1e30 -> -1e30 + ln(2); one finite -> finite (exp underflows to 0).
  // Argument of the log is in [1,2] -> always normal.
  float mx = fmaxf(a, b);
  float mn = fminf(a, b);
  return mx + fast_log(1.0f + fast_exp(mn - mx));
}

__global__ __launch_bounds__(256) void ctccrf_fwd(
    const float* __restrict__ em, const float* __restrict__ tr,
    const float* __restrict__ bos, const float* __restrict__ eos,
    const int* __restrict__ lengths, const int* __restrict__ targets,
    const int* __restrict__ tlens,
    float* __restrict__ logz_den, float* __restrict__ logz_num,
    int B, int T, int S, int L) {
  // numerator ping-pong buffers (slot 0 = the "u-1" NEG sentinel)
  __shared__ float sae[2][257];
  __shared__ float sax[2][257];
  // denominator state: alpha[16][8] double-buffered + per-batch running max
  __shared__ float sal[2][16][8];
  __shared__ float sm[16];

  if ((int)blockIdx.x < B) {
    // ------------------------- numerator lattice -------------------------
    const int b = blockIdx.x;
    const int u = threadIdx.x;
    const int Tb = lengths[b];
    const int Lb = tlens[b];
    const int base = b * T * S;  // < 2^31 for these sizes

    int es = 0, xs = 4;
    float tee = 0.f, txe = 0.f, tex = 0.f, txx = 0.f;
    float a_e = NEGV, a_x = NEGV;
    if (u < L) {
      es = targets[b * L + u];
      xs = es + 4;
      int pe = (u > 0) ? targets[b * L + u - 1] : es;
      int px = pe + 4;
      tee = tr[pe * S + es];  // enter_{u-1}  -> enter_u
      txe = tr[px * S + es];  // extend_{u-1} -> enter_u
      tex = tr[es * S + xs];  // enter_u      -> extend_u
      txx = tr[xs * S + xs];  // extend_u     -> extend_u
      if (u == 0) a_e = bos[es] + em[base + es];
    }
    if (u == 0) {
      sae[0][0] = NEGV; sax[0][0] = NEGV;
      sae[1][0] = NEGV; sax[1][0] = NEGV;
    }
    sae[0][u + 1] = a_e;
    sax[0][u + 1] = a_x;
    __syncthreads();

    // 32-bit running emission offsets, start at t=1
    int off_e = base + S + es;
    int off_x = base + S + xs;
    const float eos_e = eos[es];
    const float eos_x = eos[xs];

    for (int t = 1; t < Tb; ++t) {
      const int cur = (t - 1) & 1;
      const int nxt = t & 1;
      const float pae = sae[cur][u];  // a_e[u-1]
      const float pax = sax[cur][u];  // a_x[u-1]
      const float eme = em[off_e];
      const float emx = em[off_x];
      off_e += S;
      off_x += S;
      const float ne = eme + lae(pae + tee, pax + txe);
      const float nx = emx + lae(a_e + tex, a_x + txx);
      a_e = ne;
      a_x = nx;
      sae[nxt][u + 1] = a_e;
      sax[nxt][u + 1] = a_x;
      __syncthreads();
    }
    if (u == Lb - 1) logz_num[b] = lae(a_e + eos_e, a_x + eos_x);

  } else {
    // --------------------- denominator (WMMA, 1 wave) --------------------
    const int tid = threadIdx.x;

    // alpha0[b][s] = bos[s] + em[b,0,s]
    for (int i = tid; i < 16 * 8; i += blockDim.x) {
      const int bb = i >> 3, s = i & 7;
      sal[0][bb][s] = bos[s] + em[bb * T * S + s];
    }
    __syncthreads();

    const int len_b = (tid < 16) ? lengths[tid] : 0x7fffffff;  // preloaded

    if (tid < 16) {
      float mx = sal[0][tid][0];
#pragma unroll
      for (int s = 1; s < 8; ++s) mx = fmaxf(mx, sal[0][tid][s]);
      sm[tid] = mx;
      if (len_b == 1) {
        float m2 = sal[0][tid][0] + eos[0];
#pragma unroll
        for (int s = 1; s < 8; ++s) m2 = fmaxf(m2, sal[0][tid][s] + eos[s]);
        float acc = 0.f;
#pragma unroll
        for (int s = 0; s < 8; ++s)
          acc += fast_exp(sal[0][tid][s] + eos[s] - m2);
        logz_den[tid] = m2 + fast_log(acc);
      }
    }
    __syncthreads();

    // Per-lane WMMA layout constants (wave32):
    //  A 16x4 f32: lane m=tid&15 holds K = {kb, kb+1}, kb = 2*(tid>=16)
    //  B  4x16 f32: lane n=tid&15 holds K = {kb, kb+1} for column n
    //  C/D 16x16 f32: VGPR v on lane l is (M = v + 8*(l>=16), N = l&15)
    const int n = tid & 15;
    const int hi = tid >> 4;
    const int kb = hi * 2;
    const int n7 = n & 7;  // columns 8..15 of D are unused; clamp tr index
    v2f b1, b2;
    b1.x = fast_exp(tr[(kb + 0) * 8 + n7]);  // E[k][n] for K = 0..3
    b1.y = fast_exp(tr[(kb + 1) * 8 + n7]);
    b2.x = fast_exp(tr[(kb + 4) * 8 + n7]);  // E[k][n] for K = 4..7
    b2.y = fast_exp(tr[(kb + 5) * 8 + n7]);

    // 32-bit running emission offsets for the 8 D rows this lane extracts:
    // offv[v] -> em[bb][t=1][n], bb = v + 8*hi. Advanced by S each step.
    int offv[8];
#pragma unroll
    for (int v = 0; v < 8; ++v) offv[v] = (v + 8 * hi) * T * S + S + n;

    for (int t = 1; t < T; ++t) {
      const int cur = (t - 1) & 1;
      const int nxt = t & 1;
      if (tid < 32) {  // wave 0, full EXEC inside (uniform branch)
        const float mm = sm[n];  // n doubles as the A-row (batch) index
        v2f a1, a2;
        a1.x = fast_exp(sal[cur][n][kb + 0] - mm);
        a1.y = fast_exp(sal[cur][n][kb + 1] - mm);
        a2.x = fast_exp(sal[cur][n][kb + 4] - mm);
        a2.y = fast_exp(sal[cur][n][kb + 5] - mm);
        v8f c = {};
        // D[b][s'] = sum_{s=0..7} exp(alpha[b][s]-m_b) * exp(Tr[s][s'])
        c = __builtin_amdgcn_wmma_f32_16x16x4_f32(false, a1, false, b1,
                                                  (short)0, c, false, false);
        c = __builtin_amdgcn_wmma_f32_16x16x4_f32(false, a2, false, b2,
                                                  (short)0, c, false, false);
        if (n < 8) {
          // D >= exp(alpha_max - m)*E > 0 and never denormal -> fast_log ok.
#pragma unroll
          for (int v = 0; v < 8; ++v) {
            const int bb = v + 8 * hi;
            sal[nxt][bb][n] = fast_log(c[v]) + sm[bb] + em[offv[v]];
            offv[v] += S;
          }
        }
      }
      __syncthreads();
      if (tid < 16) {
        float mx = sal[nxt][tid][0];
#pragma unroll
        for (int s = 1; s < 8; ++s) mx = fmaxf(mx, sal[nxt][tid][s]);
        sm[tid] = mx;
        if (t == len_b - 1) {
          float m2 = sal[nxt][tid][0] + eos[0];
#pragma unroll
          for (int s = 1; s < 8; ++s)
            m2 = fmaxf(m2, sal[nxt][tid][s] + eos[s]);
          float acc = 0.f;
#pragma unroll
          for (int s = 0; s < 8; ++s)
            acc += fast_exp(sal[nxt][tid][s] + eos[s] - m2);
          logz_den[tid] = m2 + fast_log(acc);
        }
      }
      __syncthreads();
    }
  }
}

__global__ void ctccrf_reduce(const float* __restrict__ den,
                              const float* __restrict__ num,
                              float* __restrict__ out, int B) {
  if (threadIdx.x == 0 && blockIdx.x == 0) {
    float s = 0.f;
    for (int b = 0; b < B; ++b) s += den[b] - num[b];
    out[0] = s / (float)B;
  }
}

extern "C" void kernel_launch(void* const* d_in, const int* in_sizes, int n_in,
                              void* d_out, int out_size, void* d_ws,
                              size_t ws_size, hipStream_t stream) {
  (void)n_in; (void)out_size; (void)ws_size;
  const float* em   = (const float*)d_in[0];
  const float* tr   = (const float*)d_in[1];
  const float* bos  = (const float*)d_in[2];
  const float* eos  = (const float*)d_in[3];
  const int* lengths = (const int*)d_in[4];
  const int* targets = (const int*)d_in[5];
  const int* tlens   = (const int*)d_in[6];

  const int B = in_sizes[4];            // 16
  const int S = in_sizes[2];            // 8
  const int T = in_sizes[0] / (B * S);  // 2048
  const int L = in_sizes[5] / B;        // 256

  float* den = (float*)d_ws;
  float* num = den + B;

  ctccrf_fwd<<<dim3(B + 1), dim3(256), 0, stream>>>(
      em, tr, bos, eos, lengths, targets, tlens, den, num, B, T, S, L);
  ctccrf_reduce<<<dim3(1), dim3(32), 0, stream>>>(den, num, (float*)d_out, B);
}